// TemporalGNN_vanilla_two_layer_23811298689804
// MI455X (gfx1250) — compile-verified
//
#include <hip/hip_runtime.h>
#include <math.h>

typedef float v2f __attribute__((ext_vector_type(2)));
typedef float v8f __attribute__((ext_vector_type(8)));

#define C0v 2
#define PPv 12
#define H1v 32
#define H2v 64

__device__ __forceinline__ v8f vzero8() {
    v8f z = {0.f,0.f,0.f,0.f,0.f,0.f,0.f,0.f};
    return z;
}

// Exact fp32 WMMA: D(16x16) += A(16x4) * B(4x16), f32 throughout.
__device__ __forceinline__ v8f wmma4(v2f a, v2f b, v8f c) {
    return __builtin_amdgcn_wmma_f32_16x16x4_f32(false, a, false, b, (short)0, c,
                                                 false, false);
}

__device__ __forceinline__ float sigm(float x) { return 1.0f / (1.0f + __expf(-x)); }

// ---------------------------------------------------------------- utilities

__global__ void k_zero(float* p, long long n) {
    long long i = (long long)blockIdx.x * blockDim.x + threadIdx.x;
    if (i < n) p[i] = 0.0f;
}

__global__ void k_deg(const int* __restrict__ dst, int E, float* __restrict__ deg) {
    int e = blockIdx.x * blockDim.x + threadIdx.x;
    if (e < E) atomicAdd(&deg[dst[e]], 1.0f);
}

__global__ void k_dinv(float* deg, int n) {
    int i = blockIdx.x * blockDim.x + threadIdx.x;
    if (i < n) deg[i] = rsqrtf(deg[i] + 1.0f);
}

// agg[dst] += dinv[src]*dinv[dst] * x[src]   (W floats per row, W % 4 == 0)
template <int W>
__global__ void k_scatter(const int* __restrict__ src, const int* __restrict__ dst,
                          const float* __restrict__ x, const float* __restrict__ dinv,
                          float* __restrict__ agg, int E) {
    int e = blockIdx.x * blockDim.x + threadIdx.x;
    if (e >= E) return;
    int s = src[e], d = dst[e];
    float c = dinv[s] * dinv[d];
    const float4* xs = (const float4*)(x + (size_t)s * W);
    float* ad = agg + (size_t)d * W;
#pragma unroll
    for (int q = 0; q < W / 4; ++q) {
        float4 v = xs[q];
        atomicAdd(ad + q * 4 + 0, c * v.x);
        atomicAdd(ad + q * 4 + 1, c * v.y);
        atomicAdd(ad + q * 4 + 2, c * v.z);
        atomicAdd(ad + q * 4 + 3, c * v.w);
    }
}

// agg[n][*] += dinv[n]^2 * x[n][*]
__global__ void k_self(const float* __restrict__ x, const float* __restrict__ dinv,
                       float* __restrict__ agg, int n, int w) {
    long long i = (long long)blockIdx.x * blockDim.x + threadIdx.x;
    if (i < (long long)n * w) {
        int node = (int)(i / w);
        float d = dinv[node];
        agg[i] += d * d * x[i];
    }
}

// ----------------------------------------------------------------- layer 1
// h1 = relu( sum_p probs[p] * (1 - Z_p) * tanh(Ht_p) ),   H1 = 32.
// One wave per 16-node tile. cz/ch A-fragments built in registers from aggx
// (16x2 @ 2x32 is folded into fragment construction), then 16x32 @ 32x32 via
// 8-step chains of V_WMMA_F32_16X16X4_F32 per 16-wide output tile.
__global__ void __launch_bounds__(256)
k_layer1(const float* __restrict__ aggx, const float* __restrict__ W1,
         const float* __restrict__ b1, const float* __restrict__ Wl1,
         const float* __restrict__ bl1, const float* __restrict__ att,
         float* __restrict__ h1, int n) {
    int lane = threadIdx.x & 31;
    int wid  = threadIdx.x >> 5;
    int tile = blockIdx.x * 8 + wid;
    int row  = lane & 15;
    int half = lane >> 4;

    // softmax over att (12 values)
    float pr[PPv];
    float mx = -1e30f;
#pragma unroll
    for (int p = 0; p < PPv; ++p) { pr[p] = att[p]; mx = fmaxf(mx, pr[p]); }
    float ssum = 0.f;
#pragma unroll
    for (int p = 0; p < PPv; ++p) { pr[p] = __expf(pr[p] - mx); ssum += pr[p]; }
    float isum = 1.0f / ssum;
#pragma unroll
    for (int p = 0; p < PPv; ++p) pr[p] *= isum;

    int noder = tile * 16 + row;
    if (noder > n - 1) noder = n - 1;

    const float* W1z = W1;               // gate 0, (2,32)
    const float* W1h = W1 + 2 * 2 * H1v; // gate 2
    const float* b1z = b1;
    const float* b1h = b1 + 2 * H1v;
    const float* Wlz = Wl1;                    // gate 0, rows 0..31 of (64,32)
    const float* Wlh = Wl1 + 2 * 2 * H1v * H1v;

    v8f acc[2];
    acc[0] = vzero8();
    acc[1] = vzero8();

    for (int p = 0; p < PPv; ++p) {
        float ax0 = aggx[(size_t)noder * (C0v * PPv) + p];
        float ax1 = aggx[(size_t)noder * (C0v * PPv) + PPv + p];

        float az[16], ah[16];
#pragma unroll
        for (int t = 0; t < 8; ++t) {
            int j0 = 4 * t + 2 * half;
            az[2 * t + 0] = ax0 * W1z[j0]     + ax1 * W1z[H1v + j0]     + b1z[j0];
            az[2 * t + 1] = ax0 * W1z[j0 + 1] + ax1 * W1z[H1v + j0 + 1] + b1z[j0 + 1];
            ah[2 * t + 0] = ax0 * W1h[j0]     + ax1 * W1h[H1v + j0]     + b1h[j0];
            ah[2 * t + 1] = ax0 * W1h[j0 + 1] + ax1 * W1h[H1v + j0 + 1] + b1h[j0 + 1];
        }

#pragma unroll
        for (int cth = 0; cth < 2; ++cth) {
            int col = cth * 16 + row;
            v8f zc = vzero8(), hc = vzero8();
#pragma unroll
            for (int t = 0; t < 8; ++t) {
                int k0 = 4 * t + 2 * half;
                v2f av = {az[2 * t], az[2 * t + 1]};
                v2f bv = {Wlz[k0 * H1v + col], Wlz[(k0 + 1) * H1v + col]};
                zc = wmma4(av, bv, zc);
            }
#pragma unroll
            for (int t = 0; t < 8; ++t) {
                int k0 = 4 * t + 2 * half;
                v2f av = {ah[2 * t], ah[2 * t + 1]};
                v2f bv = {Wlh[k0 * H1v + col], Wlh[(k0 + 1) * H1v + col]};
                hc = wmma4(av, bv, hc);
            }
            float bz = bl1[col];
            float bh = bl1[2 * H1v + col];
#pragma unroll
            for (int r = 0; r < 8; ++r) {
                float Z  = sigm(zc[r] + bz);
                float Ht = tanhf(hc[r] + bh);
                acc[cth][r] += pr[p] * (1.0f - Z) * Ht;
            }
        }
    }

#pragma unroll
    for (int cth = 0; cth < 2; ++cth) {
#pragma unroll
        for (int r = 0; r < 8; ++r) {
            int m  = r + 8 * half;
            int nd = tile * 16 + m;
            if (nd < n) h1[(size_t)nd * H1v + cth * 16 + row] = fmaxf(acc[cth][r], 0.0f);
        }
    }
}

// ----------------------------------------------------------------- layer 2
// Per 16-node tile: base = (Ah1) @ W2[g][:32] via WMMA (once), per-period
// rank-1 time update, then cz2(16x64) @ Wl2(64x64) via WMMA with an LDS
// transpose (D-layout -> A-layout). Final MLP fused: relu(h2@lw1+lb1)@lw2+lb2.
__global__ void __launch_bounds__(256)
k_layer2(const float* __restrict__ aggh, const float* __restrict__ aggx,
         const float* __restrict__ W2, const float* __restrict__ b2,
         const float* __restrict__ Wl2, const float* __restrict__ bl2,
         const float* __restrict__ att, const float* __restrict__ lw1,
         const float* __restrict__ lb1, const float* __restrict__ lw2,
         const float* __restrict__ lb2, float* __restrict__ out, int n) {
    __shared__ float xp[8][16][68];  // per-wave 16x64 transpose tile (+pad)

    int lane = threadIdx.x & 31;
    int wid  = threadIdx.x >> 5;
    int tile = blockIdx.x * 8 + wid;
    int row  = lane & 15;
    int half = lane >> 4;

    float pr[PPv];
    float mx = -1e30f;
#pragma unroll
    for (int p = 0; p < PPv; ++p) { pr[p] = att[p]; mx = fmaxf(mx, pr[p]); }
    float ssum = 0.f;
#pragma unroll
    for (int p = 0; p < PPv; ++p) { pr[p] = __expf(pr[p] - mx); ssum += pr[p]; }
    float isum = 1.0f / ssum;
#pragma unroll
    for (int p = 0; p < PPv; ++p) pr[p] *= isum;

    int noder = tile * 16 + row;
    if (noder > n - 1) noder = n - 1;

    // A-fragments of (A h1) row  (K = 32)
    float aagg[16];
#pragma unroll
    for (int t = 0; t < 8; ++t) {
        int k0 = 4 * t + 2 * half;
        aagg[2 * t + 0] = aggh[(size_t)noder * H1v + k0];
        aagg[2 * t + 1] = aggh[(size_t)noder * H1v + k0 + 1];
    }

    const float* W2z  = W2;                    // gate 0, (33,64)
    const float* W2h  = W2 + 2 * 33 * H2v;     // gate 2
    const float* b2z  = b2;
    const float* b2h  = b2 + 2 * H2v;
    const float* Wl2z = Wl2;                   // gate 0, rows 0..63 of (128,64)
    const float* Wl2h = Wl2 + 2 * 2 * H2v * H2v;

    // base GEMM: (A h1)(16x32) @ W2[g][:32](32x64), once per tile per gate
    v8f baseZ[4], baseH[4];
#pragma unroll
    for (int cth = 0; cth < 4; ++cth) {
        int col = cth * 16 + row;
        v8f zc = vzero8(), hc = vzero8();
#pragma unroll
        for (int t = 0; t < 8; ++t) {
            int k0 = 4 * t + 2 * half;
            v2f av = {aagg[2 * t], aagg[2 * t + 1]};
            v2f bv = {W2z[k0 * H2v + col], W2z[(k0 + 1) * H2v + col]};
            zc = wmma4(av, bv, zc);
        }
#pragma unroll
        for (int t = 0; t < 8; ++t) {
            int k0 = 4 * t + 2 * half;
            v2f av = {aagg[2 * t], aagg[2 * t + 1]};
            v2f bv = {W2h[k0 * H2v + col], W2h[(k0 + 1) * H2v + col]};
            hc = wmma4(av, bv, hc);
        }
        baseZ[cth] = zc;
        baseH[cth] = hc;
    }

    // node ids for the D-layout rows this lane owns
    int ndm[8];
#pragma unroll
    for (int r = 0; r < 8; ++r) {
        int nd = tile * 16 + r + 8 * half;
        ndm[r] = nd > n - 1 ? n - 1 : nd;
    }

    v8f acc2[4];
#pragma unroll
    for (int c = 0; c < 4; ++c) acc2[c] = vzero8();

    for (int p = 0; p < PPv; ++p) {
        float tv[8];  // (A time_p) for this lane's D rows = aggx channel 1
#pragma unroll
        for (int r = 0; r < 8; ++r)
            tv[r] = aggx[(size_t)ndm[r] * (C0v * PPv) + PPv + p];

        // ---- gate Z: cz2 = base + tv * W2[g][32] + b2, staged to LDS ----
#pragma unroll
        for (int cth = 0; cth < 4; ++cth) {
            int col = cth * 16 + row;
            float wr = W2z[32 * H2v + col];
            float bb = b2z[col];
#pragma unroll
            for (int r = 0; r < 8; ++r)
                xp[wid][r + 8 * half][col] = baseZ[cth][r] + tv[r] * wr + bb;
        }
        __syncthreads();
        v8f z2[4];
#pragma unroll
        for (int cth = 0; cth < 4; ++cth) {
            int col = cth * 16 + row;
            v8f c8 = vzero8();
#pragma unroll
            for (int t = 0; t < 16; ++t) {
                int k0 = 4 * t + 2 * half;
                v2f av = {xp[wid][row][k0], xp[wid][row][k0 + 1]};
                v2f bv = {Wl2z[k0 * H2v + col], Wl2z[(k0 + 1) * H2v + col]};
                c8 = wmma4(av, bv, c8);
            }
            z2[cth] = c8;
        }
        __syncthreads();

        // ---- gate H ----
#pragma unroll
        for (int cth = 0; cth < 4; ++cth) {
            int col = cth * 16 + row;
            float wr = W2h[32 * H2v + col];
            float bb = b2h[col];
#pragma unroll
            for (int r = 0; r < 8; ++r)
                xp[wid][r + 8 * half][col] = baseH[cth][r] + tv[r] * wr + bb;
        }
        __syncthreads();
#pragma unroll
        for (int cth = 0; cth < 4; ++cth) {
            int col = cth * 16 + row;
            v8f c8 = vzero8();
#pragma unroll
            for (int t = 0; t < 16; ++t) {
                int k0 = 4 * t + 2 * half;
                v2f av = {xp[wid][row][k0], xp[wid][row][k0 + 1]};
                v2f bv = {Wl2h[k0 * H2v + col], Wl2h[(k0 + 1) * H2v + col]};
                c8 = wmma4(av, bv, c8);
            }
            float bz = bl2[col];
            float bh = bl2[2 * H2v + col];
#pragma unroll
            for (int r = 0; r < 8; ++r) {
                float Z  = sigm(z2[cth][r] + bz);
                float Ht = tanhf(c8[r] + bh);
                acc2[cth][r] += pr[p] * (1.0f - Z) * Ht;
            }
        }
        __syncthreads();
    }

    // h2 = relu(acc2) staged to LDS for the fused final MLP
#pragma unroll
    for (int cth = 0; cth < 4; ++cth) {
        int col = cth * 16 + row;
#pragma unroll
        for (int r = 0; r < 8; ++r)
            xp[wid][r + 8 * half][col] = fmaxf(acc2[cth][r], 0.0f);
    }
    __syncthreads();

    // t = relu(h2(16x64) @ lw1(64x32) + lb1) via WMMA
    v8f tt[2];
#pragma unroll
    for (int cth = 0; cth < 2; ++cth) {
        int col = cth * 16 + row;
        v8f c8 = vzero8();
#pragma unroll
        for (int t = 0; t < 16; ++t) {
            int k0 = 4 * t + 2 * half;
            v2f av = {xp[wid][row][k0], xp[wid][row][k0 + 1]};
            v2f bv = {lw1[k0 * H1v + col], lw1[(k0 + 1) * H1v + col]};
            c8 = wmma4(av, bv, c8);
        }
        tt[cth] = c8;
    }
    __syncthreads();
#pragma unroll
    for (int cth = 0; cth < 2; ++cth) {
        int col = cth * 16 + row;
        float bb = lb1[col];
#pragma unroll
        for (int r = 0; r < 8; ++r)
            xp[wid][r + 8 * half][col] = fmaxf(tt[cth][r] + bb, 0.0f);
    }
    __syncthreads();

    // final projection (16x32)@(32x12): 192 outputs, 6 per lane, scalar MACs
#pragma unroll
    for (int q = 0; q < 6; ++q) {
        int idx = lane * 6 + q;
        int m = idx / 12;
        int o = idx - m * 12;
        float v = lb2[o];
#pragma unroll
        for (int j = 0; j < H1v; ++j) v += xp[wid][m][j] * lw2[j * 12 + o];
        int nd = tile * 16 + m;
        if (nd < n) out[(size_t)nd * 12 + o] = v;
    }
}

// ----------------------------------------------------------------- launch

extern "C" void kernel_launch(void* const* d_in, const int* in_sizes, int n_in,
                              void* d_out, int out_size, void* d_ws, size_t ws_size,
                              hipStream_t stream) {
    const float* x1   = (const float*)d_in[0];
    const int*   ei   = (const int*)d_in[1];
    const float* W1   = (const float*)d_in[4];
    const float* b1   = (const float*)d_in[5];
    const float* Wl1  = (const float*)d_in[6];
    const float* bl1  = (const float*)d_in[7];
    const float* att1 = (const float*)d_in[8];
    const float* W2   = (const float*)d_in[9];
    const float* b2   = (const float*)d_in[10];
    const float* Wl2  = (const float*)d_in[11];
    const float* bl2  = (const float*)d_in[12];
    const float* att2 = (const float*)d_in[13];
    const float* lw1  = (const float*)d_in[14];
    const float* lb1  = (const float*)d_in[15];
    const float* lw2  = (const float*)d_in[16];
    const float* lb2  = (const float*)d_in[17];
    float* out = (float*)d_out;

    int n = in_sizes[0] / (C0v * PPv);
    int E = in_sizes[1] / 2;
    const int* src = ei;
    const int* dst = ei + E;

    float* ws   = (float*)d_ws;
    float* deg  = ws;                         // n        (becomes dinv)
    float* aggx = ws + n;                     // 24n      (A~ x1)
    float* h1   = aggx + (size_t)24 * n;      // 32n
    float* aggh = h1 + (size_t)32 * n;        // 32n      (A~ h1)

    long long ztot = (long long)n * (1 + 24 + 32 + 32);
    k_zero<<<(int)((ztot + 255) / 256), 256, 0, stream>>>(ws, ztot);
    k_deg<<<(E + 255) / 256, 256, 0, stream>>>(dst, E, deg);
    k_dinv<<<(n + 255) / 256, 256, 0, stream>>>(deg, n);
    k_scatter<24><<<(E + 255) / 256, 256, 0, stream>>>(src, dst, x1, deg, aggx, E);
    k_self<<<(int)(((long long)n * 24 + 255) / 256), 256, 0, stream>>>(x1, deg, aggx, n, 24);

    int tiles  = (n + 15) / 16;
    int blocks = (tiles + 7) / 8;
    k_layer1<<<blocks, 256, 0, stream>>>(aggx, W1, b1, Wl1, bl1, att1, h1, n);

    k_scatter<32><<<(E + 255) / 256, 256, 0, stream>>>(src, dst, h1, deg, aggh, E);
    k_self<<<(int)(((long long)n * 32 + 255) / 256), 256, 0, stream>>>(h1, deg, aggh, n, 32);

    k_layer2<<<blocks, 256, 0, stream>>>(aggh, aggx, W2, b2, Wl2, bl2, att2,
                                         lw1, lb1, lw2, lb2, out, n);
}